// NeighborLoss_12584254177316
// MI455X (gfx1250) — compile-verified
//
#include <hip/hip_runtime.h>
#include <math.h>

#define NDIM 16
#define NCLUS 64
#define INTRA_MARGIN 0.5f
#define INTER_2M 3.0f        // 2 * INTER_MARGIN
#define REG_W 0.001f
#define EPSV 1e-8f

// workspace layout (floats)
#define WS_SUMS  0            // 64*16
#define WS_CNT   1024         // 64
#define WS_MEANS 1088         // 64*16
#define WS_H     2112         // 64
#define WS_INTER 2176
#define WS_REG   2177
#define WS_TOTAL 2180

typedef __attribute__((ext_vector_type(2))) float v2f;
typedef __attribute__((ext_vector_type(8))) float v8f;

__global__ void k_zero(float* __restrict__ ws) {
    int i = blockIdx.x * blockDim.x + threadIdx.x;
    if (i < WS_TOTAL) ws[i] = 0.0f;
}

// Pass 1: per-cluster sums + counts, LDS-privatized then flushed with global atomics.
__global__ void k_segsum(const float* __restrict__ f, const int* __restrict__ lab,
                         float* __restrict__ ws, int n) {
    __shared__ float ls[NCLUS * NDIM];
    __shared__ float lc[NCLUS];
    for (int i = threadIdx.x; i < NCLUS * NDIM; i += blockDim.x) ls[i] = 0.0f;
    for (int i = threadIdx.x; i < NCLUS; i += blockDim.x) lc[i] = 0.0f;
    __syncthreads();

    const int stride = gridDim.x * blockDim.x;
    for (int p = blockIdx.x * blockDim.x + threadIdx.x; p < n; p += stride) {
        if (p + stride < n)
            __builtin_prefetch(f + (size_t)(p + stride) * NDIM, 0, 1);
        const int c = lab[p];
        const float4* fv = (const float4*)(f + (size_t)p * NDIM);
        float v[NDIM];
        *(float4*)(v + 0)  = fv[0];
        *(float4*)(v + 4)  = fv[1];
        *(float4*)(v + 8)  = fv[2];
        *(float4*)(v + 12) = fv[3];
        float* s = ls + c * NDIM;
#pragma unroll
        for (int k = 0; k < NDIM; ++k) atomicAdd(s + k, v[k]);
        atomicAdd(lc + c, 1.0f);
    }
    __syncthreads();
    for (int i = threadIdx.x; i < NCLUS * NDIM; i += blockDim.x)
        atomicAdd(&ws[WS_SUMS + i], ls[i]);
    for (int i = threadIdx.x; i < NCLUS; i += blockDim.x)
        atomicAdd(&ws[WS_CNT + i], lc[i]);
}

// Single workgroup: means, Gram matrix via V_WMMA_F32_16X16X4_F32,
// inter-cluster hinge loss + centroid-norm regularizer.
__global__ void __launch_bounds__(512) k_centroids(float* __restrict__ ws) {
    __shared__ float sm[NCLUS * NDIM];
    __shared__ float q[NCLUS], srow[NCLUS], rg[NCLUS];
    __shared__ float interAcc;
    const int t = threadIdx.x;
    if (t == 0) interAcc = 0.0f;

    for (int i = t; i < NCLUS * NDIM; i += 512) {
        int c = i >> 4;
        float cnt = ws[WS_CNT + c];
        float m = cnt > 0.0f ? ws[WS_SUMS + i] / cnt : 0.0f;
        sm[i] = m;
        ws[WS_MEANS + i] = m;
    }
    __syncthreads();

    if (t < NCLUS) {
        float qq = 0.0f, ss = 0.0f, rr = 0.0f;
#pragma unroll
        for (int k = 0; k < NDIM; ++k) {
            float m = sm[t * NDIM + k];
            qq += m * m;
            ss += m;
            float me = m + EPSV;
            rr += me * me;
        }
        q[t] = qq; srow[t] = ss; rg[t] = sqrtf(rr);
    }
    __syncthreads();

    // 16 waves -> 16 tiles of the 64x64 Gram matrix, 4 chained K-steps each.
    const int wave = t >> 5, lane = t & 31;
    const int ib = (wave >> 2) * 16, jb = (wave & 3) * 16;
    const int r = lane & 15, hi = lane >> 4;

    v8f acc = {};
#pragma unroll
    for (int kc = 0; kc < 4; ++kc) {
        const int k0 = kc * 4 + hi * 2;
        v2f A, B;
        A.x = sm[(ib + r) * NDIM + k0];
        A.y = sm[(ib + r) * NDIM + k0 + 1];
        B.x = sm[(jb + r) * NDIM + k0];
        B.y = sm[(jb + r) * NDIM + k0 + 1];
        acc = __builtin_amdgcn_wmma_f32_16x16x4_f32(
            false, A, false, B, (short)0, acc, false, false);
    }

    float h = 0.0f;
#pragma unroll
    for (int vr = 0; vr < 8; ++vr) {
        const int i = ib + vr + 8 * hi;
        const int j = jb + r;
        if (i != j) {
            float g = acc[vr];
            // exact ||m_i - m_j + eps||^2 expansion
            float d2 = q[i] + q[j] - 2.0f * g
                     + 2.0f * EPSV * (srow[i] - srow[j])
                     + (float)NDIM * EPSV * EPSV;
            d2 = d2 > 0.0f ? d2 : 0.0f;
            float dd = sqrtf(d2);
            float tt = INTER_2M - dd;
            tt = tt > 0.0f ? tt : 0.0f;
            h += tt * tt;
        }
    }
    atomicAdd(&interAcc, h);
    __syncthreads();

    if (t == 0) {
        ws[WS_INTER] = interAcc / (float)(NCLUS * (NCLUS - 1));
        float rs = 0.0f;
        for (int c = 0; c < NCLUS; ++c) rs += rg[c];
        ws[WS_REG] = rs / (float)NCLUS;
    }
}

// Pass 2: per-point hinge^2 against own centroid, per-cluster LDS accumulation.
__global__ void k_intra(const float* __restrict__ f, const int* __restrict__ lab,
                        float* __restrict__ ws, int n) {
    __shared__ float sm[NCLUS * NDIM];
    __shared__ float lh[NCLUS];
    for (int i = threadIdx.x; i < NCLUS * NDIM; i += blockDim.x) sm[i] = ws[WS_MEANS + i];
    for (int i = threadIdx.x; i < NCLUS; i += blockDim.x) lh[i] = 0.0f;
    __syncthreads();

    const int stride = gridDim.x * blockDim.x;
    for (int p = blockIdx.x * blockDim.x + threadIdx.x; p < n; p += stride) {
        if (p + stride < n)
            __builtin_prefetch(f + (size_t)(p + stride) * NDIM, 0, 1);
        const int c = lab[p];
        const float4* fv = (const float4*)(f + (size_t)p * NDIM);
        float v[NDIM];
        *(float4*)(v + 0)  = fv[0];
        *(float4*)(v + 4)  = fv[1];
        *(float4*)(v + 8)  = fv[2];
        *(float4*)(v + 12) = fv[3];
        const float* m = sm + c * NDIM;
        float d2 = 0.0f;
#pragma unroll
        for (int k = 0; k < NDIM; ++k) {
            float e = v[k] - m[k] + EPSV;
            d2 += e * e;
        }
        float dist = sqrtf(d2);
        float tt = dist - INTRA_MARGIN;
        tt = tt > 0.0f ? tt : 0.0f;
        atomicAdd(&lh[c], tt * tt);
    }
    __syncthreads();
    for (int i = threadIdx.x; i < NCLUS; i += blockDim.x)
        atomicAdd(&ws[WS_H + i], lh[i]);
}

__global__ void k_final(const float* __restrict__ ws, float* __restrict__ out) {
    if (threadIdx.x == 0 && blockIdx.x == 0) {
        float intra = 0.0f;
        for (int c = 0; c < NCLUS; ++c) {
            float cnt = ws[WS_CNT + c];
            intra += cnt > 0.0f ? ws[WS_H + c] / cnt : 0.0f;
        }
        intra /= (float)NCLUS;
        out[0] = intra + ws[WS_INTER] + REG_W * ws[WS_REG];
    }
}

extern "C" void kernel_launch(void* const* d_in, const int* in_sizes, int n_in,
                              void* d_out, int out_size, void* d_ws, size_t ws_size,
                              hipStream_t stream) {
    const float* feat = (const float*)d_in[0];
    const int*   lab  = (const int*)d_in[1];
    float* ws  = (float*)d_ws;
    float* out = (float*)d_out;
    const int n = in_sizes[1];  // N points (labels count)

    k_zero<<<(WS_TOTAL + 255) / 256, 256, 0, stream>>>(ws);
    k_segsum<<<1024, 256, 0, stream>>>(feat, lab, ws, n);
    k_centroids<<<1, 512, 0, stream>>>(ws);
    k_intra<<<1024, 256, 0, stream>>>(feat, lab, ws, n);
    k_final<<<1, 64, 0, stream>>>(ws, out);
}